// ComplexBinauralExtraction_38963943309891
// MI455X (gfx1250) — compile-verified
//
#include <hip/hip_runtime.h>
#include <hip/hip_bf16.h>
#include <math.h>

// ---------------- problem constants (from reference) ----------------
constexpr int kB  = 4;
constexpr int kE  = 512;
constexpr int kC  = 8;
constexpr int kT  = 501;
constexpr int kD  = 512;
constexpr int kHR = 256;
constexpr int kH  = 8;
constexpr int kCE = kC * kE;      // 4096
constexpr int kROWS = kB * kT;    // 2004
constexpr int kHD = kD / kH;      // 64
constexpr int kHDH = kHR / kH;    // 32

typedef float v2f __attribute__((ext_vector_type(2)));
typedef float v8f __attribute__((ext_vector_type(8)));

// ---------------- WMMA f32 16x16x4 wrapper (CDNA5) ----------------
__device__ __forceinline__ v8f wmma4(v2f a, v2f b, v8f c) {
  return __builtin_amdgcn_wmma_f32_16x16x4_f32(false, a, false, b, (short)0, c,
                                               false, false);
}

// Tiling for the big GEMM-like kernels
constexpr int GBM = 16;    // output rows per block
constexpr int GBN = 128;   // output cols per block (8 waves x 16)
constexpr int GBK = 32;    // K per LDS stage
constexpr int GLS = GBK + 2;  // padded LDS row stride (floats, keeps parity even)

// =====================================================================
// Generic complex linear: Y[r,n] = sum_k X[r,k]*W[n,k] + bias[n]
// X: (rows,K) complex interleaved; W: (N,K) complex; Y: (rows,N) complex.
// Block: 256 threads = 8 waves; each wave owns one 16x16 complex tile.
// Requires: K % 32 == 0, N % 128 == 0, rows >= 1 (row-clamped loads).
// =====================================================================
__global__ __launch_bounds__(256)
void cgemm_linear(const float* __restrict__ X, const float* __restrict__ W,
                  const float* __restrict__ Bv, float* __restrict__ Y,
                  int rows, int K, int N) {
  __shared__ float sXre[GBM][GLS], sXim[GBM][GLS];
  __shared__ float sWre[GBN][GLS], sWim[GBN][GLS];
  const int tid  = threadIdx.x;
  const int lane = tid & 31, wv = tid >> 5;
  const int half = lane >> 4, l15 = lane & 15;
  const int r0 = blockIdx.y * GBM;
  const int n0 = blockIdx.x * GBN;

  v8f arr{}, aii{}, ari{}, air{};
  for (int k0 = 0; k0 < K; k0 += GBK) {
    __syncthreads();
    {
      // X tile: 16 rows x 16 complex-pairs -> one float4 per thread
      const int p = tid & 15, row = tid >> 4;
      const int rr = min(r0 + row, rows - 1);      // clamp: unused rows never stored
      const float4 xa = *(const float4*)(X + ((size_t)rr * K + k0 + p * 2) * 2);
      sXre[row][p * 2] = xa.x; sXre[row][p * 2 + 1] = xa.z;
      sXim[row][p * 2] = xa.y; sXim[row][p * 2 + 1] = xa.w;
      // W tile: 128 n x 16 complex-pairs -> 8 float4 per thread
#pragma unroll
      for (int i = 0; i < 8; ++i) {
        const int idx = tid + i * 256;
        const int pp = idx & 15, nn = idx >> 4;
        const float4 wb = *(const float4*)(W + ((size_t)(n0 + nn) * K + k0 + pp * 2) * 2);
        sWre[nn][pp * 2] = wb.x; sWre[nn][pp * 2 + 1] = wb.z;
        sWim[nn][pp * 2] = wb.y; sWim[nn][pp * 2 + 1] = wb.w;
      }
    }
    __syncthreads();
#pragma unroll
    for (int kk = 0; kk < GBK; kk += 4) {
      const int kA = kk + 2 * half;
      const v2f axr = *(const v2f*)&sXre[l15][kA];
      const v2f axi = *(const v2f*)&sXim[l15][kA];
      const v2f bwr = *(const v2f*)&sWre[wv * 16 + l15][kA];
      const v2f bwi = *(const v2f*)&sWim[wv * 16 + l15][kA];
      arr = wmma4(axr, bwr, arr);
      aii = wmma4(axi, bwi, aii);
      ari = wmma4(axr, bwi, ari);
      air = wmma4(axi, bwr, air);
    }
  }
  const int col = n0 + wv * 16 + l15;              // C/D: lane&15 == N
  const float br = Bv ? Bv[2 * col] : 0.f;
  const float bi = Bv ? Bv[2 * col + 1] : 0.f;
  const v8f orr = arr - aii;
  const v8f oim = ari + air;
#pragma unroll
  for (int v = 0; v < 8; ++v) {
    const int gr = r0 + v + half * 8;              // C/D: M = v + 8*(lane>>4)
    if (gr < rows) {
      float* p = Y + ((size_t)gr * N + col) * 2;
      p[0] = orr[v] + br;
      p[1] = oim[v] + bi;
    }
  }
}

// =====================================================================
// Attention scores: Wout[b,h,q,k] = crelu(scale*(Q_h . K_h^T) + Mask)
// Q,Km: (B,T,D) complex; Wout/Mask: (B,H,T,T) complex.
// Block: 256 threads, tile 16 q-rows x 128 k-tokens, K over hd in steps of 32.
// Clamped loads on both sides (stores guarded by q<T && k<T).
// =====================================================================
__global__ __launch_bounds__(256)
void attn_scores(const float* __restrict__ Q, const float* __restrict__ Km,
                 const float* __restrict__ Mask, float* __restrict__ Wout,
                 int T, int D, int H, float scale) {
  __shared__ float sXre[GBM][GLS], sXim[GBM][GLS];
  __shared__ float sWre[GBN][GLS], sWim[GBN][GLS];
  const int bh = blockIdx.z, b = bh / H, h = bh % H;
  const int hd = D / H;
  const int tid  = threadIdx.x;
  const int lane = tid & 31, wv = tid >> 5;
  const int half = lane >> 4, l15 = lane & 15;
  const int q0 = blockIdx.y * GBM;
  const int c0 = blockIdx.x * GBN;

  v8f arr{}, aii{}, ari{}, air{};
  for (int k0 = 0; k0 < hd; k0 += GBK) {
    __syncthreads();
    {
      const int p = tid & 15, row = tid >> 4;
      const int rq = min(q0 + row, T - 1);
      const float4 xa =
          *(const float4*)(Q + (((size_t)b * T + rq) * D + h * hd + k0 + p * 2) * 2);
      sXre[row][p * 2] = xa.x; sXre[row][p * 2 + 1] = xa.z;
      sXim[row][p * 2] = xa.y; sXim[row][p * 2 + 1] = xa.w;
#pragma unroll
      for (int i = 0; i < 8; ++i) {
        const int idx = tid + i * 256;
        const int pp = idx & 15, nn = idx >> 4;
        const int tk = min(c0 + nn, T - 1);
        const float4 wb =
            *(const float4*)(Km + (((size_t)b * T + tk) * D + h * hd + k0 + pp * 2) * 2);
        sWre[nn][pp * 2] = wb.x; sWre[nn][pp * 2 + 1] = wb.z;
        sWim[nn][pp * 2] = wb.y; sWim[nn][pp * 2 + 1] = wb.w;
      }
    }
    __syncthreads();
#pragma unroll
    for (int kk = 0; kk < GBK; kk += 4) {
      const int kA = kk + 2 * half;
      const v2f axr = *(const v2f*)&sXre[l15][kA];
      const v2f axi = *(const v2f*)&sXim[l15][kA];
      const v2f bwr = *(const v2f*)&sWre[wv * 16 + l15][kA];
      const v2f bwi = *(const v2f*)&sWim[wv * 16 + l15][kA];
      arr = wmma4(axr, bwr, arr);
      aii = wmma4(axi, bwi, aii);
      ari = wmma4(axr, bwi, ari);
      air = wmma4(axi, bwr, air);
    }
  }
  const int gc = c0 + wv * 16 + l15;
#pragma unroll
  for (int v = 0; v < 8; ++v) {
    const int gq = q0 + v + half * 8;
    if (gq < T && gc < T) {
      float sr = (arr[v] - aii[v]) * scale;
      float si = (ari[v] + air[v]) * scale;
      const size_t mi = ((((size_t)b * H + h) * T + gq) * T + gc) * 2;
      if (Mask) { sr += Mask[mi]; si += Mask[mi + 1]; }
      Wout[mi]     = fmaxf(sr, 0.f);
      Wout[mi + 1] = fmaxf(si, 0.f);
    }
  }
}

// =====================================================================
// Attention apply: O[b,q,h*hd+d] = sum_k Wm[b,h,q,k] * V[b,k,h*hd+d]
// Block: 128 threads = 4 waves covering the full hd=64 columns.
// K loop over T padded to a multiple of 32; mask (A) zero-filled past T
// via value-select (no exec branch); V clamped (A==0 kills the product).
// V staged transposed in LDS so B fragments are contiguous v2f.
// =====================================================================
__global__ __launch_bounds__(128)
void attn_apply(const float* __restrict__ Wm, const float* __restrict__ V,
                float* __restrict__ O, int T, int D, int H) {
  __shared__ float sAre[16][GLS], sAim[16][GLS];
  __shared__ float sBre[64][GLS], sBim[64][GLS];
  const int bh = blockIdx.y, b = bh / H, h = bh % H;
  const int hd = D / H;   // 64
  const int tid  = threadIdx.x;
  const int lane = tid & 31, wv = tid >> 5;   // 0..3
  const int half = lane >> 4, l15 = lane & 15;
  const int q0 = blockIdx.x * GBM;
  const int Kpad = ((T + GBK - 1) / GBK) * GBK;

  v8f arr{}, aii{}, ari{}, air{};
  for (int k0 = 0; k0 < Kpad; k0 += GBK) {
    __syncthreads();
    // A tile: 16 q x 32 k complex -> 4 float2 per thread
#pragma unroll
    for (int i = 0; i < 4; ++i) {
      const int idx = tid + i * 128;
      const int k = idx & 31, qq = idx >> 5;
      const int rq = min(q0 + qq, T - 1);
      const int kg = k0 + k;
      float2 a = *(const float2*)(Wm +
                 ((((size_t)b * H + h) * T + rq) * T + min(kg, T - 1)) * 2);
      if (kg >= T) a = make_float2(0.f, 0.f);   // value select, no exec branch
      sAre[qq][k] = a.x; sAim[qq][k] = a.y;
    }
    // B tile (transposed): 64 d x 32 k complex -> 16 float2 per thread
#pragma unroll
    for (int i = 0; i < 16; ++i) {
      const int idx = tid + i * 128;
      const int nn = idx & 63, kk = idx >> 6;   // kk 0..31
      const int tk = min(k0 + kk, T - 1);
      const float2 vv =
          *(const float2*)(V + (((size_t)b * T + tk) * D + h * hd + nn) * 2);
      sBre[nn][kk] = vv.x; sBim[nn][kk] = vv.y;
    }
    __syncthreads();
#pragma unroll
    for (int kk = 0; kk < GBK; kk += 4) {
      const int kA = kk + 2 * half;
      const v2f axr = *(const v2f*)&sAre[l15][kA];
      const v2f axi = *(const v2f*)&sAim[l15][kA];
      const v2f bwr = *(const v2f*)&sBre[wv * 16 + l15][kA];
      const v2f bwi = *(const v2f*)&sBim[wv * 16 + l15][kA];
      arr = wmma4(axr, bwr, arr);
      aii = wmma4(axi, bwi, aii);
      ari = wmma4(axr, bwi, ari);
      air = wmma4(axi, bwr, air);
    }
  }
  const int col = h * hd + wv * 16 + l15;
#pragma unroll
  for (int v = 0; v < 8; ++v) {
    const int gq = q0 + v + half * 8;
    if (gq < T) {
      float* p = O + (((size_t)b * T + gq) * D + col) * 2;
      p[0] = arr[v] - aii[v];
      p[1] = ari[v] + air[v];
    }
  }
}

// =====================================================================
// Tiny HRTF attention core: 2 tokens, hd=32, per (b,h) scalar math.
// =====================================================================
__global__ void hrtf_attn_core(const float* __restrict__ Q, const float* __restrict__ K,
                               const float* __restrict__ V, float* __restrict__ O) {
  const int b = blockIdx.x;
  const int h = threadIdx.x;
  if (h >= kH) return;
  const float scale = rsqrtf((float)kHDH);
  float wr[2][2], wi[2][2];
  for (int qi = 0; qi < 2; ++qi)
    for (int ki = 0; ki < 2; ++ki) {
      float sr = 0.f, si = 0.f;
      for (int d = 0; d < kHDH; ++d) {
        const size_t qo = (((size_t)b * 2 + qi) * kHR + h * kHDH + d) * 2;
        const size_t ko = (((size_t)b * 2 + ki) * kHR + h * kHDH + d) * 2;
        const float qr = Q[qo], qii = Q[qo + 1];
        const float kr = K[ko], kii = K[ko + 1];
        sr += qr * kr - qii * kii;
        si += qr * kii + qii * kr;
      }
      wr[qi][ki] = fmaxf(sr * scale, 0.f);
      wi[qi][ki] = fmaxf(si * scale, 0.f);
    }
  for (int qi = 0; qi < 2; ++qi)
    for (int d = 0; d < kHDH; ++d) {
      float orr = 0.f, oim = 0.f;
      for (int ki = 0; ki < 2; ++ki) {
        const size_t vo = (((size_t)b * 2 + ki) * kHR + h * kHDH + d) * 2;
        const float vr = V[vo], vi = V[vo + 1];
        orr += wr[qi][ki] * vr - wi[qi][ki] * vi;
        oim += wr[qi][ki] * vi + wi[qi][ki] * vr;
      }
      const size_t oo = (((size_t)b * 2 + qi) * kHR + h * kHDH + d) * 2;
      O[oo] = orr; O[oo + 1] = oim;
    }
}

// =====================================================================
// Complex "global layer norm": magnitude norm, keep phase, complex
// per-channel gamma/beta. One block per batch element, no atomics.
// =====================================================================
__global__ __launch_bounds__(1024)
void gln_kernel(const float* __restrict__ X, float* __restrict__ Y,
                const float* __restrict__ G, const float* __restrict__ Bt,
                int nch, int inner) {
  const int b = blockIdx.x;
  const size_t N = (size_t)nch * inner;
  const float* xb = X + (size_t)b * N * 2;
  float* yb = Y + (size_t)b * N * 2;
  __shared__ float s1[1024];
  __shared__ float s2[1024];
  float a1 = 0.f, a2 = 0.f;
  for (size_t i = threadIdx.x; i < N; i += blockDim.x) {
    const float re = xb[2 * i], im = xb[2 * i + 1];
    const float m = sqrtf(re * re + im * im);
    a1 += m; a2 += m * m;
  }
  s1[threadIdx.x] = a1; s2[threadIdx.x] = a2;
  __syncthreads();
  for (int s = 512; s > 0; s >>= 1) {
    if ((int)threadIdx.x < s) {
      s1[threadIdx.x] += s1[threadIdx.x + s];
      s2[threadIdx.x] += s2[threadIdx.x + s];
    }
    __syncthreads();
  }
  const float mean = s1[0] / (float)N;
  const float var  = s2[0] / (float)N - mean * mean;
  const float inv  = rsqrtf(var + 1e-8f);
  for (size_t i = threadIdx.x; i < N; i += blockDim.x) {
    const float re = xb[2 * i], im = xb[2 * i + 1];
    const float m = sqrtf(re * re + im * im);
    const float nm = (m - mean) * inv;
    float nr, ni;
    if (m > 1e-30f) { const float f = nm / m; nr = re * f; ni = im * f; }
    else            { nr = nm; ni = 0.f; }
    const int c = (int)(i / inner);
    const float gr = G[2 * c], gi = G[2 * c + 1];
    const float br = Bt[2 * c], bi = Bt[2 * c + 1];
    yb[2 * i]     = nr * gr - ni * gi + br;
    yb[2 * i + 1] = nr * gi + ni * gr + bi;
  }
}

// =====================================================================
// Elementwise helpers
// =====================================================================
__global__ void pack_x0(const float* __restrict__ Em, float* __restrict__ X0) {
  const size_t i = (size_t)blockIdx.x * blockDim.x + threadIdx.x;
  const size_t total = (size_t)kB * kT * kC * kE;
  if (i >= total) return;
  size_t r = i;
  const int e = (int)(r % kE); r /= kE;
  const int c = (int)(r % kC); r /= kC;
  const int t = (int)(r % kT); r /= kT;
  const int b = (int)r;
  const size_t src = ((((size_t)b * kE + e) * kC + c) * kT + t) * 2;
  X0[2 * i] = Em[src]; X0[2 * i + 1] = Em[src + 1];
}

__global__ void bmul_kernel(float* __restrict__ X, const float* __restrict__ Hv) {
  const size_t i = (size_t)blockIdx.x * blockDim.x + threadIdx.x;
  const size_t total = (size_t)kB * kT * kD;
  if (i >= total) return;
  const int d = (int)(i % kD);
  const int b = (int)(i / ((size_t)kT * kD));
  const float xr = X[2 * i], xi = X[2 * i + 1];
  const float hr = Hv[((size_t)b * kD + d) * 2], hi = Hv[((size_t)b * kD + d) * 2 + 1];
  X[2 * i]     = xr * hr - xi * hi;
  X[2 * i + 1] = xr * hi + xi * hr;
}

__global__ void permute_out(const float* __restrict__ Y, float* __restrict__ Out) {
  const size_t i = (size_t)blockIdx.x * blockDim.x + threadIdx.x;
  const size_t total = (size_t)kB * kE * kC * kT;
  if (i >= total) return;
  size_t r = i;
  const int t = (int)(r % kT); r /= kT;
  const int c = (int)(r % kC); r /= kC;
  const int e = (int)(r % kE); r /= kE;
  const int b = (int)r;
  const size_t src = (((size_t)b * kT + t) * (size_t)(kE * kC) + (size_t)e * kC + c) * 2;
  Out[2 * i] = Y[src]; Out[2 * i + 1] = Y[src + 1];
}

// =====================================================================
// Host-side orchestration
// =====================================================================
extern "C" void kernel_launch(void* const* d_in, const int* in_sizes, int n_in,
                              void* d_out, int out_size, void* d_ws, size_t ws_size,
                              hipStream_t stream) {
  (void)in_sizes; (void)n_in; (void)out_size; (void)ws_size;
  const float* emb  = (const float*)d_in[0];
  const float* hrtf = (const float*)d_in[1];
  int ip = 2;
  auto nxt = [&](void) { return (const float*)d_in[ip++]; };
  const float *fc1_w = nxt(), *fc1_b = nxt();
  const float *mx_qw = nxt(), *mx_qb = nxt(), *mx_kw = nxt(), *mx_kb = nxt();
  const float *mx_vw = nxt(), *mx_vb = nxt(), *mx_ow = nxt(), *mx_ob = nxt();
  const float *hh_qw = nxt(), *hh_qb = nxt(), *hh_kw = nxt(), *hh_kb = nxt();
  const float *hh_vw = nxt(), *hh_vb = nxt(), *hh_ow = nxt(), *hh_ob = nxt();
  const float *glnh_g = nxt(), *glnh_b = nxt();
  const float *fch_w = nxt(), *fch_b = nxt();
  const float* sa_w[4][8]; const float *g_g[4], *g_b[4];
  for (int i = 0; i < 4; ++i) {
    for (int j = 0; j < 8; ++j) sa_w[i][j] = nxt();
    g_g[i] = nxt(); g_b[i] = nxt();
  }
  const float *out_w = nxt(), *out_b = nxt();

  // ---- workspace layout (floats) ----
  float* ws = (float*)d_ws;
  const size_t tokBIG = (size_t)kB * kT * kCE * 2;
  const size_t tokD   = (size_t)kB * kT * kD * 2;
  const size_t maskSz = (size_t)kB * kH * kT * kT * 2;
  size_t off = 0;
  float* BIG = ws + off; off += tokBIG;
  float* Xa  = ws + off; off += tokD;
  float* Qa  = ws + off; off += tokD;
  float* Ka  = ws + off; off += tokD;
  float* Va  = ws + off; off += tokD;
  float* AO  = ws + off; off += tokD;
  float* Ya  = ws + off; off += tokD;
  float* M0  = ws + off; off += maskSz;
  float* M1  = ws + off; off += maskSz;
  const size_t hsz = (size_t)kB * 2 * kHR * 2;
  float* HQ = ws + off; off += hsz;
  float* HK = ws + off; off += hsz;
  float* HV = ws + off; off += hsz;
  float* HO = ws + off; off += hsz;
  float* HP = ws + off; off += hsz;
  float* HN = ws + off; off += hsz;
  float* H2 = ws + off; off += (size_t)kB * kD * 2;

  const int rowT16 = (kROWS + 15) / 16;              // 126
  const int t16    = (kT + 15) / 16;                 // 32
  const int t128   = (kT + 127) / 128;               // 4
  const float scaleD = 0.125f;                       // 64^-0.5

  // 1) pack emb -> x0
  {
    const size_t tot = (size_t)kB * kT * kC * kE;
    pack_x0<<<dim3((unsigned)((tot + 255) / 256)), 256, 0, stream>>>(emb, BIG);
  }
  // 2) fc1: (2004 x 4096) -> (2004 x 512)
  cgemm_linear<<<dim3(kD / GBN, rowT16), 256, 0, stream>>>(BIG, fc1_w, fc1_b, Xa,
                                                           kROWS, kCE, kD);
  // 3) sa_mix attention
  cgemm_linear<<<dim3(kD / GBN, rowT16), 256, 0, stream>>>(Xa, mx_qw, mx_qb, Qa, kROWS, kD, kD);
  cgemm_linear<<<dim3(kD / GBN, rowT16), 256, 0, stream>>>(Xa, mx_kw, mx_kb, Ka, kROWS, kD, kD);
  cgemm_linear<<<dim3(kD / GBN, rowT16), 256, 0, stream>>>(Xa, mx_vw, mx_vb, Va, kROWS, kD, kD);
  attn_scores<<<dim3(t128, t16, kB * kH), 256, 0, stream>>>(Qa, Ka, nullptr, M0, kT, kD, kH, scaleD);
  attn_apply<<<dim3(t16, kB * kH), 128, 0, stream>>>(M0, Va, AO, kT, kD, kH);
  cgemm_linear<<<dim3(kD / GBN, rowT16), 256, 0, stream>>>(AO, mx_ow, mx_ob, Xa, kROWS, kD, kD);

  // 4) HRTF path: rows = B*2 = 8, d = 256
  cgemm_linear<<<dim3(kHR / GBN, 1), 256, 0, stream>>>(hrtf, hh_qw, hh_qb, HQ, kB * 2, kHR, kHR);
  cgemm_linear<<<dim3(kHR / GBN, 1), 256, 0, stream>>>(hrtf, hh_kw, hh_kb, HK, kB * 2, kHR, kHR);
  cgemm_linear<<<dim3(kHR / GBN, 1), 256, 0, stream>>>(hrtf, hh_vw, hh_vb, HV, kB * 2, kHR, kHR);
  hrtf_attn_core<<<dim3(kB), kH, 0, stream>>>(HQ, HK, HV, HO);
  cgemm_linear<<<dim3(kHR / GBN, 1), 256, 0, stream>>>(HO, hh_ow, hh_ob, HP, kB * 2, kHR, kHR);
  gln_kernel<<<dim3(kB), 1024, 0, stream>>>(HP, HN, glnh_g, glnh_b, 2, kHR);
  cgemm_linear<<<dim3(kD / GBN, 1), 256, 0, stream>>>(HN, fch_w, fch_b, H2, kB, 2 * kHR, kD);

  // 5) bn = x * h (broadcast complex multiply, in place)
  {
    const size_t tot = (size_t)kB * kT * kD;
    bmul_kernel<<<dim3((unsigned)((tot + 255) / 256)), 256, 0, stream>>>(Xa, H2);
  }

  // 6) four masked attention + gln blocks
  float* masks[2] = {M0, M1};
  const float* mask_in = nullptr;
  for (int i = 0; i < 4; ++i) {
    float* mask_out = masks[i & 1];
    cgemm_linear<<<dim3(kD / GBN, rowT16), 256, 0, stream>>>(Xa, sa_w[i][0], sa_w[i][1], Qa, kROWS, kD, kD);
    cgemm_linear<<<dim3(kD / GBN, rowT16), 256, 0, stream>>>(Xa, sa_w[i][2], sa_w[i][3], Ka, kROWS, kD, kD);
    cgemm_linear<<<dim3(kD / GBN, rowT16), 256, 0, stream>>>(Xa, sa_w[i][4], sa_w[i][5], Va, kROWS, kD, kD);
    attn_scores<<<dim3(t128, t16, kB * kH), 256, 0, stream>>>(Qa, Ka, mask_in, mask_out, kT, kD, kH, scaleD);
    attn_apply<<<dim3(t16, kB * kH), 128, 0, stream>>>(mask_out, Va, AO, kT, kD, kH);
    cgemm_linear<<<dim3(kD / GBN, rowT16), 256, 0, stream>>>(AO, sa_w[i][6], sa_w[i][7], Ya, kROWS, kD, kD);
    gln_kernel<<<dim3(kB), 1024, 0, stream>>>(Ya, Xa, g_g[i], g_b[i], kT, kD);
    mask_in = mask_out;
  }

  // 7) output projection 512 -> 4096 (reuse BIG)
  cgemm_linear<<<dim3(kCE / GBN, rowT16), 256, 0, stream>>>(Xa, out_w, out_b, BIG, kROWS, kD, kCE);

  // 8) permute to (B,E,C,T)
  {
    const size_t tot = (size_t)kB * kE * kC * kT;
    permute_out<<<dim3((unsigned)((tot + 255) / 256)), 256, 0, stream>>>(BIG, (float*)d_out);
  }
}